// YOLOLoss_88003879895262
// MI455X (gfx1250) — compile-verified
//
#include <hip/hip_runtime.h>

#define GRID_H   52
#define GRID_W   52
#define GCELLS   2704            // 52*52
#define NANCH    9
#define NBATCH   256
#define NGROUP   (NBATCH * GCELLS * NANCH)   // 6,230,016 (divisible by 4)
#define NCHUNK   (NGROUP / 4)                // 1,557,504 chunks of 4 groups (80B, 16B-aligned)
#define NBLK1    1024                        // partial-sum blocks (divisible by 64 for WMMA reduce)
#define NTHR1    256

typedef float v2f __attribute__((ext_vector_type(2)));
typedef float v4f __attribute__((ext_vector_type(4)));
typedef float v8f __attribute__((ext_vector_type(8)));

__device__ __forceinline__ float fsigmoid(float x) {
    return 1.0f / (1.0f + __expf(-x));
}

// ---------------------------------------------------------------------------
// Kernel 1: streaming no-obj reduction over all (b,g,a) groups.
//   S1 = sum (pred[:4] - anchors_grid)^2   (cell terms cancel analytically)
//   S2 = sum conf^2
// Writes per-block partials: ws[blk] = S1part, ws[NBLK1+blk] = S2part.
// ---------------------------------------------------------------------------
__global__ __launch_bounds__(NTHR1) void yolo_noobj_reduce(
    const float* __restrict__ pred, const float* __restrict__ anchors,
    float* __restrict__ partials)
{
    __shared__ float s_aw2[NANCH];
    __shared__ float s_ah2[NANCH];
    __shared__ float s_red[2 * (NTHR1 / 32)];

    const int tid = threadIdx.x;
    if (tid < NANCH) {
        float aw = anchors[2 * tid + 0];
        float ah = anchors[2 * tid + 1];
        s_aw2[tid] = aw * aw;
        s_ah2[tid] = ah * ah;
    }
    __syncthreads();

    const float inv_g2 = 1.0f / (52.0f * 52.0f);
    float s1 = 0.0f, s2 = 0.0f;

    const unsigned stride = gridDim.x * blockDim.x;
    for (unsigned c = blockIdx.x * blockDim.x + tid; c < NCHUNK; c += stride) {
        const v4f* p = (const v4f*)(pred + (size_t)c * 20u);   // 80B, 16B-aligned
        v4f vv0 = __builtin_nontemporal_load(p + 0);
        v4f vv1 = __builtin_nontemporal_load(p + 1);
        v4f vv2 = __builtin_nontemporal_load(p + 2);
        v4f vv3 = __builtin_nontemporal_load(p + 3);
        v4f vv4 = __builtin_nontemporal_load(p + 4);

        unsigned cn = c + stride;                               // stream prefetch
        if (cn < NCHUNK) __builtin_prefetch(pred + (size_t)cn * 20u, 0, 0);

        float f[20];
        {
            v4f vv[5] = {vv0, vv1, vv2, vv3, vv4};
#pragma unroll
            for (int q = 0; q < 5; ++q) {
                f[4 * q + 0] = vv[q][0];
                f[4 * q + 1] = vv[q][1];
                f[4 * q + 2] = vv[q][2];
                f[4 * q + 3] = vv[q][3];
            }
        }

#pragma unroll
        for (int j = 0; j < 4; ++j) {
            float p0 = f[5 * j + 0];
            float p1 = f[5 * j + 1];
            float p2 = f[5 * j + 2];
            float p3 = f[5 * j + 3];
            float p4 = f[5 * j + 4];
            unsigned a = (4u * c + (unsigned)j) % 9u;

            float sx = fsigmoid(p0) - 0.5f;          // (x - agx)*52
            float sy = fsigmoid(p1) - 0.5f;
            float ew = __expf(p2) - 1.0f;            // (w - aw)/aw
            float eh = __expf(p3) - 1.0f;
            float sc = fsigmoid(p4);

            s1 += (sx * sx + sy * sy) * inv_g2
                +  s_aw2[a] * ew * ew + s_ah2[a] * eh * eh;
            s2 += sc * sc;
        }
    }

    // deterministic wave32 reduce, then block reduce
#pragma unroll
    for (int off = 16; off; off >>= 1) {
        s1 += __shfl_down(s1, off, 32);
        s2 += __shfl_down(s2, off, 32);
    }
    const int wave = tid >> 5;
    const int lane = tid & 31;
    if (lane == 0) { s_red[2 * wave] = s1; s_red[2 * wave + 1] = s2; }
    __syncthreads();
    if (tid == 0) {
        float t1 = 0.0f, t2 = 0.0f;
#pragma unroll
        for (int w = 0; w < NTHR1 / 32; ++w) { t1 += s_red[2 * w]; t2 += s_red[2 * w + 1]; }
        partials[blockIdx.x] = t1;
        partials[NBLK1 + blockIdx.x] = t2;
    }
}

// ---------------------------------------------------------------------------
// Kernel 2 (single block, 256 threads): per-sample obj terms + WMMA-based
// reduction of the 2*NBLK1 partials + final scalar combine.
// WMMA trick: ones(16x4) x B(4x16) + C  =>  every D cell accumulates column
// sums; the grand total over all D cells is layout-agnostic (= sum of all B
// elements fed in), so Sum(lane c[0]) over 32 lanes = 2 * total.
// ---------------------------------------------------------------------------
__global__ __launch_bounds__(256) void yolo_obj_finalize(
    const float* __restrict__ pred, const float* __restrict__ label,
    const float* __restrict__ anchors, const float* __restrict__ partials,
    float* __restrict__ out)
{
    __shared__ float r0[256], r1[256], r2[256], r3[256];
    __shared__ float s12[2];

    const int b = threadIdx.x;

    float lx = label[4 * b + 0], ly = label[4 * b + 1];
    float lw = label[4 * b + 2], lh = label[4 * b + 3];

    // responsible cell (flat = x*GH + y)
    float fx = floorf(lx * (float)GRID_W);
    float fy = floorf(ly * (float)GRID_H);
    int idx = (int)fx * GRID_H + (int)fy;

    // argmax of squared distance (first max wins, like jnp.argmax)
    int am = 0; float best = -1.0f;
#pragma unroll
    for (int a = 0; a < NANCH; ++a) {
        float dw = lw - anchors[2 * a + 0];
        float dh = lh - anchors[2 * a + 1];
        float d = dw * dw + dh * dh;
        if (d > best) { best = d; am = a; }
    }
    float aw = anchors[2 * am + 0], ah = anchors[2 * am + 1];

    const float* pg = pred + (((size_t)b * GCELLS + (size_t)idx) * NANCH + (size_t)am) * 5u;
    float p0 = pg[0], p1 = pg[1], p2 = pg[2], p3 = pg[3], p4 = pg[4];

    float x = (fsigmoid(p0) + fx) * (1.0f / 52.0f);
    float y = (fsigmoid(p1) + fy) * (1.0f / 52.0f);
    float w = aw * __expf(p2);
    float h = ah * __expf(p3);
    float conf = fsigmoid(p4);

    float agx = (fx + 0.5f) * (1.0f / 52.0f);
    float agy = (fy + 0.5f) * (1.0f / 52.0f);
    float d0 = x - agx, d1 = y - agy, d2 = w - aw, d3 = h - ah;
    float coor_sq_obj = d0 * d0 + d1 * d1 + d2 * d2 + d3 * d3;
    float conf_sq_obj = conf * conf;

    // clipped corners
    float lx1 = fmaxf(lx - lw * 0.5f, 0.0f), ly1 = fmaxf(ly - lh * 0.5f, 0.0f);
    float lx2 = fminf(lx + lw * 0.5f, 1.0f), ly2 = fminf(ly + lh * 0.5f, 1.0f);
    float px1 = fmaxf(x - w * 0.5f, 0.0f),   py1 = fmaxf(y - h * 0.5f, 0.0f);
    float px2 = fminf(x + w * 0.5f, 1.0f),   py2 = fminf(y + h * 0.5f, 1.0f);

    float ix1 = fmaxf(lx1, px1), iy1 = fmaxf(ly1, py1);
    float ix2 = fminf(lx2, px2), iy2 = fminf(ly2, py2);
    float la = lx2 * ly2, pa = px2 * py2;                 // faithful to reference
    float inter = fmaxf(ix2 - ix1, 0.0f) * fmaxf(iy2 - iy1, 0.0f);
    float iou = inter / (la + pa - inter);

    float e0 = x - lx, e1 = y - ly, e2 = w - lw, e3 = h - lh;
    float coor_obj = e0 * e0 + e1 * e1 + e2 * e2 + e3 * e3;
    float dc = conf - iou;
    float conf_obj = dc * dc;

    r0[b] = coor_sq_obj; r1[b] = conf_sq_obj; r2[b] = coor_obj; r3[b] = conf_obj;
    __syncthreads();
#pragma unroll
    for (int off = 128; off; off >>= 1) {
        if (b < off) {
            r0[b] += r0[b + off]; r1[b] += r1[b + off];
            r2[b] += r2[b + off]; r3[b] += r3[b + off];
        }
        __syncthreads();
    }

    // --- WMMA reduction of the no-obj partials (wave 0 only, EXEC all-ones) ---
    if (b < 32) {
        v2f ones; ones[0] = 1.0f; ones[1] = 1.0f;
        v8f c1 = {}; v8f c2 = {};
        const int lane = b;
        for (int base = 0; base < NBLK1; base += 64) {
            v2f b1; b1[0] = partials[base + lane];          b1[1] = partials[base + 32 + lane];
            c1 = __builtin_amdgcn_wmma_f32_16x16x4_f32(false, ones, false, b1,
                                                       (short)0, c1, false, false);
            v2f b2; b2[0] = partials[NBLK1 + base + lane];  b2[1] = partials[NBLK1 + base + 32 + lane];
            c2 = __builtin_amdgcn_wmma_f32_16x16x4_f32(false, ones, false, b2,
                                                       (short)0, c2, false, false);
        }
        float t1 = c1[0], t2 = c2[0];
#pragma unroll
        for (int off = 16; off; off >>= 1) {
            t1 += __shfl_xor(t1, off, 32);
            t2 += __shfl_xor(t2, off, 32);
        }
        if (lane == 0) { s12[0] = 0.5f * t1; s12[1] = 0.5f * t2; }
    }
    __syncthreads();

    if (b == 0) {
        float S1 = s12[0], S2 = s12[1];
        const float GA1 = (float)(GCELLS * NANCH - 1);      // 24335
        float coor_l_noobj = (S1 - r0[0]) / ((float)NBATCH * GA1 * 4.0f);
        float conf_l_noobj = (S2 - r1[0]) / ((float)NBATCH * GA1);
        float coor_l_obj   = r2[0] * (1.0f / (float)(NBATCH * 4));
        float conf_l_obj   = r3[0] * (1.0f / (float)NBATCH);
        out[0] = coor_l_obj + coor_l_noobj + conf_l_obj + conf_l_noobj;
    }
}

extern "C" void kernel_launch(void* const* d_in, const int* in_sizes, int n_in,
                              void* d_out, int out_size, void* d_ws, size_t ws_size,
                              hipStream_t stream)
{
    (void)in_sizes; (void)n_in; (void)out_size; (void)ws_size;
    const float* pred    = (const float*)d_in[0];
    const float* label   = (const float*)d_in[1];
    const float* anchors = (const float*)d_in[2];
    float* out      = (float*)d_out;
    float* partials = (float*)d_ws;   // 2*NBLK1 floats = 8 KB scratch

    yolo_noobj_reduce<<<NBLK1, NTHR1, 0, stream>>>(pred, anchors, partials);
    yolo_obj_finalize<<<1, 256, 0, stream>>>(pred, label, anchors, partials, out);
}